// BatchGraphAttention_21646635172523
// MI455X (gfx1250) — compile-verified
//
#include <hip/hip_runtime.h>

typedef float v2f __attribute__((ext_vector_type(2)));
typedef float v8f __attribute__((ext_vector_type(8)));

#define B_    8
#define N_    2048
#define F_IN  64
#define H_    4
#define F_OUT 32
#define LOG2E 1.44269504088896340736f

// ---------------------------------------------------------------------------
// Phase 1: feats = X @ kernel per head (packed layout [b][h][j/4][f][j%4]),
//          a_s = feats . attn_self, a_n = feats . attn_neigh,
//          asmax[b][h] = max(0, max_j a_s)  (int-punned atomicMax, 0-floored)
// ---------------------------------------------------------------------------
__global__ __launch_bounds__(128)
void gat_phase1(const float* __restrict__ X, const float* __restrict__ Wk,
                const float* __restrict__ attn_self, const float* __restrict__ attn_neigh,
                float* __restrict__ featsP, float* __restrict__ a_s,
                float* __restrict__ a_n, float* __restrict__ asmax)
{
    int blk  = blockIdx.x;          // B * N/8 blocks
    int b    = blk >> 8;            // 256 blocks per batch
    int n0   = (blk & 255) << 3;    // 8 rows per block
    int t    = threadIdx.x;
    int h    = t >> 5;              // wave id == head
    int f    = t & 31;              // lane == output feature
    const float* Xb  = X  + (size_t)b * N_ * F_IN;
    const float* Kh  = Wk + (size_t)h * F_IN * F_OUT;
    float as_w = attn_self [h * F_OUT + f];
    float an_w = attn_neigh[h * F_OUT + f];
    float localmax = -1e30f;

    for (int r = 0; r < 8; ++r) {
        int n = n0 + r;
        const float* Xn = Xb + (size_t)n * F_IN;
        float acc = 0.f;
#pragma unroll
        for (int c = 0; c < F_IN; ++c)
            acc = fmaf(Xn[c], Kh[c * F_OUT + f], acc);  // X broadcast, Wk coalesced

        // packed feats: [b][h][n/4][f][n%4]  -> B-fragment is one b64 load
        featsP[((((size_t)(b*H_ + h) * (N_/4) + (n >> 2)) * F_OUT + f) << 2) + (n & 3)] = acc;

        // wave32 cross-lane reductions for the two attention dots
        float u = acc * as_w;
        float w = acc * an_w;
#pragma unroll
        for (int m = 16; m >= 1; m >>= 1) {
            u += __shfl_xor(u, m, 32);
            w += __shfl_xor(w, m, 32);
        }
        if (f == 0) {
            a_s[(size_t)(b*H_ + h) * N_ + n] = u;
            a_n[(size_t)(b*H_ + h) * N_ + n] = w;
            localmax = fmaxf(localmax, u);
        }
    }
    // nonneg float compare == int compare; slot was memset to 0.0f
    if (f == 0 && localmax > 0.f)
        atomicMax((int*)(asmax + b*H_ + h), __float_as_int(localmax));
}

// ---------------------------------------------------------------------------
// Phase 2: stream A once; fused mask + LeakyReLU + softmax (fixed upper-bound
// shift c_i) + P x feats via V_WMMA_F32_16X16X4_F32; all 4 heads per A tile.
// Block = (b, 16-row tile), 8 waves split the j range, LDS-atomic reduce.
// ---------------------------------------------------------------------------
__global__ __launch_bounds__(256)
void gat_phase2(const float* __restrict__ A, const float* __restrict__ bias,
                const float* __restrict__ featsP, const float* __restrict__ a_s,
                const float* __restrict__ a_n, const float* __restrict__ asmax,
                float* __restrict__ out)
{
    __shared__ float lds_acc[H_][16][F_OUT];   // 8 KB
    __shared__ float lds_l  [H_][16];

    int blk  = blockIdx.x;            // B * N/16
    int b    = blk >> 7;
    int i0   = (blk & 127) << 4;
    int t    = threadIdx.x;
    int w    = t >> 5;                // wave 0..7
    int lane = t & 31;
    int m15  = lane & 15;             // A-frag row / C-frag column
    int koff = (lane >> 4) << 1;      // 0 or 2: K offset for this lane half

    // zero LDS accumulators
    {
        float* lf = &lds_acc[0][0][0];
        for (int i = t; i < H_*16*F_OUT; i += 256) lf[i] = 0.f;
        float* ll = &lds_l[0][0];
        for (int i = t; i < H_*16; i += 256) ll[i] = 0.f;
    }
    __syncthreads();

    // per-head, per-row constants: c_i = leaky(asmax + a_n[i]) >= all scores
    float an[H_], ci[H_];
    const float* asb[H_];
    const float* fpb[H_];
#pragma unroll
    for (int h = 0; h < H_; ++h) {
        float mx = asmax[b*H_ + h];
        float a  = a_n[(size_t)(b*H_ + h) * N_ + i0 + m15];
        an[h] = a;
        float s = mx + a;
        ci[h] = fmaxf(s, 0.2f * s);
        asb[h] = a_s    + (size_t)(b*H_ + h) * N_;
        fpb[h] = featsP + (size_t)(b*H_ + h) * (N_/4) * (F_OUT*4);
    }

    const float* Arow = A + ((size_t)b * N_ + (i0 + m15)) * N_;

    v8f zero = {0.f,0.f,0.f,0.f,0.f,0.f,0.f,0.f};
    v8f acc[H_][2];
#pragma unroll
    for (int h = 0; h < H_; ++h) { acc[h][0] = zero; acc[h][1] = zero; }
    float lpart[H_] = {0.f, 0.f, 0.f, 0.f};

    // 8 waves cover 32 j's per superstep; wave w owns cols j0..j0+3
    for (int s = 0; s < N_/32; ++s) {
        int j0 = (w << 2) + (s << 5);
        v2f av = *(const v2f*)(Arow + j0 + koff);       // A in A-frag layout
#pragma unroll
        for (int h = 0; h < H_; ++h) {
            v2f asv = *(const v2f*)(asb[h] + j0 + koff);
            float s0 = asv.x + an[h];  s0 = fmaxf(s0, 0.2f * s0);   // LeakyReLU(0.2)
            float s1 = asv.y + an[h];  s1 = fmaxf(s1, 0.2f * s1);
            float p0 = (av.x > 0.5f) ? __builtin_amdgcn_exp2f((s0 - ci[h]) * LOG2E) : 0.f;
            float p1 = (av.y > 0.5f) ? __builtin_amdgcn_exp2f((s1 - ci[h]) * LOG2E) : 0.f;
            lpart[h] += p0 + p1;
            v2f afrag; afrag.x = p0; afrag.y = p1;      // 16x4 A-matrix fragment
            const float* fp = fpb[h] + (size_t)(j0 >> 2) * (F_OUT*4);
#pragma unroll
            for (int half = 0; half < 2; ++half) {
                // 4x16 B-matrix fragment: lane half selects K{0,1}/K{2,3}
                v2f bfrag = *(const v2f*)(fp + (((half << 4) + m15) << 2) + koff);
                acc[h][half] = __builtin_amdgcn_wmma_f32_16x16x4_f32(
                    false, afrag, false, bfrag, (short)0, acc[h][half], false, false);
            }
        }
    }

    // reduce C fragments + l across the 8 waves via LDS atomics
    int roff = (lane >> 4) << 3;   // C-frag: rows r (lanes 0-15) / r+8 (lanes 16-31)
#pragma unroll
    for (int h = 0; h < H_; ++h) {
        atomicAdd(&lds_l[h][m15], lpart[h]);
#pragma unroll
        for (int half = 0; half < 2; ++half)
#pragma unroll
            for (int r = 0; r < 8; ++r)
                atomicAdd(&lds_acc[h][r + roff][(half << 4) + m15], acc[h][half][r]);
    }
    __syncthreads();

    // finalize: divide by softmax denom, +bias, ReLU, coalesced store [B,N,H*F_]
    int col = t & 127;
    int hh  = col >> 5;
    int ff  = col & 31;
    float bv = bias[hh * F_OUT + ff];
    int r0 = (t >> 7) << 3;
    for (int r = r0; r < r0 + 8; ++r) {
        float v = lds_acc[hh][r][ff] / lds_l[hh][r] + bv;
        out[((size_t)b * N_ + i0 + r) * (H_ * F_OUT) + col] = fmaxf(v, 0.f);
    }
}

// ---------------------------------------------------------------------------
extern "C" void kernel_launch(void* const* d_in, const int* in_sizes, int n_in,
                              void* d_out, int out_size, void* d_ws, size_t ws_size,
                              hipStream_t stream)
{
    const float* X          = (const float*)d_in[0];
    const float* A          = (const float*)d_in[1];
    const float* Wk         = (const float*)d_in[2];
    const float* bias       = (const float*)d_in[3];
    const float* attn_self  = (const float*)d_in[4];
    const float* attn_neigh = (const float*)d_in[5];
    float* out = (float*)d_out;

    // workspace layout (floats): featsP | a_s | a_n | asmax  (~8.9 MB)
    float* featsP = (float*)d_ws;
    float* a_s    = featsP + (size_t)B_ * H_ * N_ * F_OUT;
    float* a_n    = a_s    + (size_t)B_ * H_ * N_;
    float* asmax  = a_n    + (size_t)B_ * H_ * N_;

    hipMemsetAsync(asmax, 0, B_ * H_ * sizeof(float), stream);  // 0.0f floor for max
    gat_phase1<<<B_ * (N_/8), 128, 0, stream>>>(X, Wk, attn_self, attn_neigh,
                                                featsP, a_s, a_n, asmax);
    gat_phase2<<<B_ * (N_/16), 256, 0, stream>>>(A, bias, featsP, a_s, a_n, asmax, out);
}